// LTCPredictor_11725260718177
// MI455X (gfx1250) — compile-verified
//
#include <hip/hip_runtime.h>
#include <math.h>

#define ODE_UNFOLDS 6
#define EPS_F 1e-8f
#define BB 256
#define TT 128
#define NIN 64
#define NU 256
#define GG 4              // batches per block

// ---- native tanh on gfx1250 (TRANS pipe op). Builtin preferred, inline asm fallback. ----
__device__ __forceinline__ float tanh_hw(float x) {
#if __has_builtin(__builtin_amdgcn_tanhf)
  return __builtin_amdgcn_tanhf(x);
#elif __has_builtin(__builtin_amdgcn_tanh_f32)
  return __builtin_amdgcn_tanh_f32(x);
#else
  float r;
  asm("v_tanh_f32 %0, %1" : "=v"(r) : "v"(x));
  return r;
#endif
}

// gate decomposition: sigmoid(s*(v-m)) = 0.5 + 0.5*tanh(0.5*s*v - 0.5*s*m)
// contributions: w*gate = 0.5*w + (0.5*w)*t ; w*erev*gate = 0.5*w*erev + (0.5*w*erev)*t
// the 0.5*sum(...) constants are folded into per-column Knum/Kden.
__device__ __forceinline__ float gate_t(float A, float C, float x) {
  return tanh_hw(fmaf(A, x, C));
}

// ---- async global->LDS (gfx1250). Guarded; falls back to a plain load+store. ----
#if __has_builtin(__builtin_amdgcn_global_load_async_to_lds_b32)
#define HAVE_ASYNC_LDS 1
#else
#define HAVE_ASYNC_LDS 0
#endif

__device__ __forceinline__ void async_ld_b32(const float* gp, float* lp) {
#if HAVE_ASYNC_LDS
  typedef __attribute__((address_space(1))) int GI;   // global
  typedef __attribute__((address_space(3))) int LI;   // LDS
  __builtin_amdgcn_global_load_async_to_lds_b32((GI*)gp, (LI*)lp, 0, 0);
#else
  *lp = *gp;
#endif
}

__device__ __forceinline__ void async_wait0() {
#if HAVE_ASYNC_LDS
#if __has_builtin(__builtin_amdgcn_s_wait_asynccnt)
  __builtin_amdgcn_s_wait_asynccnt(0);
#else
  asm volatile("s_wait_asynccnt 0" ::: "memory");
#endif
#endif
}

// ---- pack recurrent params: {A,C} float2 + we float (erev is exactly +/-1 so 0.5*w = |we|) ----
__global__ void pack_rec(const float* __restrict__ mu, const float* __restrict__ sigma,
                         const float* __restrict__ w, const float* __restrict__ erev,
                         float2* __restrict__ Pac, float* __restrict__ Pwe) {
  int idx = blockIdx.x * blockDim.x + threadIdx.x;       // 0 .. 65535
  float s = sigma[idx], m = mu[idx], ww = w[idx], e = erev[idx];
  Pac[idx] = make_float2(0.5f * s, -0.5f * s * m);
  Pwe[idx] = 0.5f * ww * e;
}

// ---- pack sensory params, folding input_w/input_b into the affine gate argument ----
__global__ void pack_sen(const float* __restrict__ mu, const float* __restrict__ sigma,
                         const float* __restrict__ w, const float* __restrict__ erev,
                         const float* __restrict__ iw, const float* __restrict__ ib,
                         float2* __restrict__ Pac, float* __restrict__ Pwe) {
  int idx = blockIdx.x * blockDim.x + threadIdx.x;       // 0 .. 16383
  int is = idx >> 8;                                     // pre index (IN)
  float s = sigma[idx], m = mu[idx], ww = w[idx], e = erev[idx];
  float wi = iw[is], bi = ib[is];
  Pac[idx] = make_float2(0.5f * s * wi, 0.5f * s * (bi - m));
  Pwe[idx] = 0.5f * ww * e;
}

// ---- per-column constants: gleak*vleak + 0.5*sum(w*erev), gleak+EPS+0.5*sum(w), cm*6 ----
__global__ void col_consts(const float* __restrict__ PweR, const float* __restrict__ PweS,
                           const float* __restrict__ gleak, const float* __restrict__ vleak,
                           const float* __restrict__ cm,
                           float* __restrict__ Knum, float* __restrict__ Kden,
                           float* __restrict__ cm6) {
  int j = threadIdx.x;                                   // 0..255
  float sn = 0.f, sd = 0.f;
  for (int i = 0; i < NU; ++i)  { float p = PweR[i * NU + j]; sn += p; sd += fabsf(p); }
  for (int i = 0; i < NIN; ++i) { float p = PweS[i * NU + j]; sn += p; sd += fabsf(p); }
  Knum[j] = gleak[j] * vleak[j] + sn;
  Kden[j] = gleak[j] + EPS_F + sd;
  cm6[j]  = cm[j] * (float)ODE_UNFOLDS;
}

// ---- main persistent LTC kernel: block = 4 batches x 256 post-neurons, loops all T*6 steps ----
__launch_bounds__(256)
__global__ void ltc_main(const float2* __restrict__ PacR, const float* __restrict__ PweR,
                         const float2* __restrict__ PacS, const float* __restrict__ PweS,
                         const float* __restrict__ Knum, const float* __restrict__ Kden,
                         const float* __restrict__ cm6,
                         const float* __restrict__ x, const float* __restrict__ ts,
                         const float* __restrict__ ow, const float* __restrict__ ob,
                         float* __restrict__ hbuf) {
  const int j  = threadIdx.x;          // post-neuron
  const int b0 = blockIdx.x * GG;      // first batch of this block
  const int g_ld = j >> 6;             // this thread's slot in the x-slab load
  const int is_ld = j & 63;

  __shared__ float v[2][GG][NU];
  __shared__ float xs[2][GG][NIN];

  #pragma unroll
  for (int g = 0; g < GG; ++g) v[0][g][j] = 0.0f;

  const float kn  = Knum[j];
  const float kd  = Kden[j];
  const float c6  = cm6[j];
  const float owj = ow[j];
  const float obj = ob[j];

  // synchronous preload of timestep 0 inputs
  xs[0][g_ld][is_ld] = x[((size_t)(b0 + g_ld) * TT + 0) * NIN + is_ld];

  int cur = 0;
  int xb = 0;
  for (int t = 0; t < TT; ++t) {
    __syncthreads();   // xs[xb] (and v[cur]) visible to all waves

    // async prefetch of next timestep's inputs into the other x buffer
    if (t + 1 < TT)
      async_ld_b32(&x[((size_t)(b0 + g_ld) * TT + (t + 1)) * NIN + is_ld],
                   &xs[xb ^ 1][g_ld][is_ld]);

    // ---- sensory sums (x fixed within timestep) ----
    float sn[GG], sd[GG];
    #pragma unroll
    for (int g = 0; g < GG; ++g) { sn[g] = 0.f; sd[g] = 0.f; }
    #pragma unroll 4
    for (int is = 0; is < NIN; ++is) {
      float2 ac = PacS[is * NU + j];
      float  we = PweS[is * NU + j];
      float  aw = fabsf(we);
      #pragma unroll
      for (int g = 0; g < GG; ++g) {
        float gt = gate_t(ac.x, ac.y, xs[xb][g][is]);
        sn[g] = fmaf(we, gt, sn[g]);
        sd[g] = fmaf(aw, gt, sd[g]);
      }
    }

    float cmt[GG];
    #pragma unroll
    for (int g = 0; g < GG; ++g) cmt[g] = c6 / ts[(size_t)(b0 + g) * TT + t];

    // ---- 6 semi-implicit Euler unfolds ----
    for (int u = 0; u < ODE_UNFOLDS; ++u) {
      float num[GG], den[GG];
      #pragma unroll
      for (int g = 0; g < GG; ++g) { num[g] = 0.f; den[g] = 0.f; }
      #pragma unroll 4
      for (int i = 0; i < NU; ++i) {
        float2 ac = PacR[i * NU + j];      // coalesced over j, L2-resident stream
        float  we = PweR[i * NU + j];
        float  aw = fabsf(we);
        #pragma unroll
        for (int g = 0; g < GG; ++g) {
          float gt = gate_t(ac.x, ac.y, v[cur][g][i]);   // LDS broadcast
          num[g] = fmaf(we, gt, num[g]);
          den[g] = fmaf(aw, gt, den[g]);
        }
      }
      int nxt = cur ^ 1;
      #pragma unroll
      for (int g = 0; g < GG; ++g) {
        float vj   = v[cur][g][j];
        float nume = fmaf(cmt[g], vj, kn + sn[g] + num[g]);
        float dene = cmt[g] + kd + sd[g] + den[g];
        v[nxt][g][j] = nume / dene;
      }
      __syncthreads();
      cur = nxt;
    }

    // make sure this wave's async prefetch has landed before the loop-top barrier
    if (t + 1 < TT) async_wait0();
    xb ^= 1;
  }

  // output mapping of last step -> h buffer for the WMMA head
  #pragma unroll
  for (int g = 0; g < GG; ++g)
    hbuf[(size_t)(b0 + g) * NU + j] = fmaf(v[cur][g][j], owj, obj);
}

// ---- head: out[b] = sum_j h[b,j]*hw[j] + hb  via V_WMMA_F32_16X16X4_F32 ----
typedef float v2f __attribute__((ext_vector_type(2)));
typedef float v8f __attribute__((ext_vector_type(8)));

__global__ void head_wmma(const float* __restrict__ h, const float* __restrict__ hw,
                          const float* __restrict__ hb, float* __restrict__ out) {
  const int lane = threadIdx.x;            // 32 lanes, EXEC all ones
  const int bt   = blockIdx.x * 16;        // batch tile
  const int m    = lane & 15;              // A row (batch within tile)
  const int ks   = (lane >> 4) << 1;       // lanes 0-15 hold K=0,1 ; lanes 16-31 hold K=2,3

  v8f acc = {0.f, 0.f, 0.f, 0.f, 0.f, 0.f, 0.f, 0.f};
  for (int k0 = 0; k0 < NU; k0 += 4) {
    v2f a, bv;
    const float* hr = h + (size_t)(bt + m) * NU + k0 + ks;
    a[0]  = hr[0];
    a[1]  = hr[1];
    bv[0] = hw[k0 + ks];       // B[k][n] = hw[k] replicated across n
    bv[1] = hw[k0 + ks + 1];
    acc = __builtin_amdgcn_wmma_f32_16x16x4_f32(false, a, false, bv,
                                                (short)0, acc, false, false);
  }
  // D col 0: lane 0 vgprs 0..7 = rows 0..7 ; lane 16 vgprs 0..7 = rows 8..15
  float bias = hb[0];
  if (lane == 0) {
    #pragma unroll
    for (int r = 0; r < 8; ++r) out[bt + r] = acc[r] + bias;
  } else if (lane == 16) {
    #pragma unroll
    for (int r = 0; r < 8; ++r) out[bt + 8 + r] = acc[r] + bias;
  }
}

extern "C" void kernel_launch(void* const* d_in, const int* in_sizes, int n_in,
                              void* d_out, int out_size, void* d_ws, size_t ws_size,
                              hipStream_t stream) {
  const float* x    = (const float*)d_in[0];
  const float* ts   = (const float*)d_in[1];
  const float* smu  = (const float*)d_in[2];
  const float* ssig = (const float*)d_in[3];
  const float* sw   = (const float*)d_in[4];
  const float* ser  = (const float*)d_in[5];
  const float* mu   = (const float*)d_in[6];
  const float* sig  = (const float*)d_in[7];
  const float* w    = (const float*)d_in[8];
  const float* er   = (const float*)d_in[9];
  const float* gl   = (const float*)d_in[10];
  const float* vl   = (const float*)d_in[11];
  const float* cm   = (const float*)d_in[12];
  const float* iw   = (const float*)d_in[13];
  const float* ib   = (const float*)d_in[14];
  const float* ow   = (const float*)d_in[15];
  const float* obv  = (const float*)d_in[16];
  const float* hw   = (const float*)d_in[17];
  const float* hb   = (const float*)d_in[18];
  float* out = (float*)d_out;

  char* ws = (char*)d_ws;
  size_t off = 0;
  auto alloc = [&](size_t bytes) -> void* {
    void* p = ws + off;
    off += bytes;
    off = (off + 255) & ~(size_t)255;
    return p;
  };

  float2* PacR = (float2*)alloc((size_t)NU * NU * sizeof(float2));
  float*  PweR = (float*) alloc((size_t)NU * NU * sizeof(float));
  float2* PacS = (float2*)alloc((size_t)NIN * NU * sizeof(float2));
  float*  PweS = (float*) alloc((size_t)NIN * NU * sizeof(float));
  float*  Knum = (float*) alloc(NU * sizeof(float));
  float*  Kden = (float*) alloc(NU * sizeof(float));
  float*  cm6  = (float*) alloc(NU * sizeof(float));
  float*  hbuf = (float*) alloc((size_t)BB * NU * sizeof(float));

  pack_rec<<<(NU * NU) / 256, 256, 0, stream>>>(mu, sig, w, er, PacR, PweR);
  pack_sen<<<(NIN * NU) / 256, 256, 0, stream>>>(smu, ssig, sw, ser, iw, ib, PacS, PweS);
  col_consts<<<1, 256, 0, stream>>>(PweR, PweS, gl, vl, cm, Knum, Kden, cm6);
  ltc_main<<<BB / GG, 256, 0, stream>>>(PacR, PweR, PacS, PweS, Knum, Kden, cm6,
                                        x, ts, ow, obv, hbuf);
  head_wmma<<<BB / 16, 32, 0, stream>>>(hbuf, hw, hb, out);
}